// Grnet_tensors_24429773979693
// MI455X (gfx1250) — compile-verified
//
#include <hip/hip_runtime.h>
#include <math.h>

// ---------------------------------------------------------------------------
// GrNet forward: FRMap (batched GEMM, fp32 WMMA) -> ReOrth (MGS QR in LDS)
//                -> FC + log_softmax.
// Target: gfx1250 (CDNA5, wave32, WMMA, 320KB LDS/WGP).
// ---------------------------------------------------------------------------

typedef float v2f __attribute__((ext_vector_type(2)));
typedef float v8f __attribute__((ext_vector_type(8)));

#define BATCH 64
#define D0 400
#define NN 200          // N of each matrix
#define NF 16
#define D1 300
#define NCLS 10
#define FC_IN (NF * D1 * NN)   // 960000
#define MT ((D1 + 15) / 16)    // 19 M-tiles (304 >= 300)

static __device__ __forceinline__ int imin(int a, int b) { return a < b ? a : b; }

// ---------------------------------------------------------------------------
// Kernel 1: X1[b,f] = filters[f] (300x400) @ x[b] (400x200), fp32 WMMA.
// grid = (19, 1024); block = 128 (4 waves). Wave w computes N-tiles [4w,4w+3]
// of one M-tile, reusing the A fragment across 4 WMMAs per K-step.
//
// Edge handling: load offsets are CLAMPED in-bounds (garbage rows/cols of the
// fragments only affect C entries that are never stored), so the K-loop is
// branch-free. All fragment accesses are SGPR-base + 32-bit VGPR offset so
// they lower to global_load (GVS mode) rather than flat_load: no 64-bit
// per-lane pointer math, no DScnt coupling.
// ---------------------------------------------------------------------------
__global__ __launch_bounds__(128)
void frmap_wmma(const float* __restrict__ x, const float* __restrict__ filters,
                float* __restrict__ X1)
{
    const int mt   = blockIdx.x;        // 0..18
    const int bf   = blockIdx.y;        // 0..1023  (b*16 + f)
    const int b    = bf >> 4;
    const int f    = bf & 15;
    const int lane = threadIdx.x & 31;
    const int wave = threadIdx.x >> 5;

    const float* A = filters + (size_t)f * D1 * D0;  // row-major [300][400]
    const float* B = x       + (size_t)b * D0 * NN;  // row-major [400][200]
    float*       C = X1      + (size_t)bf * D1 * NN; // row-major [300][200]

    const int row   = imin(mt * 16 + (lane & 15), D1 - 1);  // clamped A row
    const int kHalf = (lane >> 4) * 2;                      // K sub-offset

    int col[4];          // true output columns (for stores)
    int boff[4];         // clamped 32-bit element offsets into B
    v8f acc[4] = {};
#pragma unroll
    for (int q = 0; q < 4; ++q) {
        col[q]  = (wave * 4 + q) * 16 + (lane & 15);
        boff[q] = kHalf * NN + imin(col[q], NN - 1);
    }
    int aoff = row * D0 + kHalf;

#pragma unroll 4
    for (int k = 0; k < D0; k += 4) {
        // Prefetch the next K-chunk of the B panel toward the WGP.
        __builtin_prefetch(B + boff[0] + 4 * NN, 0, 3);

        // A fragment 16x4: VGPR0 = K {0|2}, VGPR1 = K {1|3} per lane half
        const v2f a = *(const v2f*)(A + aoff);
        aoff += 4;
#pragma unroll
        for (int q = 0; q < 4; ++q) {
            // B fragment 4x16: VGPR0 = row (k+kHalf), VGPR1 = row (k+kHalf+1)
            v2f bb;
            bb.x = B[boff[q]];
            bb.y = B[boff[q] + NN];
            boff[q] += 4 * NN;
            // (neg_a, A, neg_b, B, c_mod, C, reuse_a, reuse_b)
            acc[q] = __builtin_amdgcn_wmma_f32_16x16x4_f32(
                false, a, false, bb, (short)0, acc[q], false, false);
        }
    }

    // C/D layout: VGPR r -> M = mt*16 + (lane<16 ? r : 8+r), N = tile + lane&15
    const int mbase = mt * 16 + (lane >> 4) * 8;
#pragma unroll
    for (int q = 0; q < 4; ++q) {
        if (col[q] >= NN) continue;
#pragma unroll
        for (int r = 0; r < 8; ++r) {
            const int m = mbase + r;
            if (m < D1) C[(size_t)m * NN + col[q]] = acc[q][r];
        }
    }
}

// ---------------------------------------------------------------------------
// Kernel 2: in-place reduced QR (Q only) via Modified Gram-Schmidt, entirely
// in LDS. One block per (b,f) matrix; 240 KB matrix + 1 KB scratch in LDS
// (CDNA5: 320 KB/WGP). 256 threads = 8 waves; wave-per-column dot/axpy.
// ---------------------------------------------------------------------------
__global__ __launch_bounds__(256)
void mgs_qr(float* __restrict__ X1)
{
    extern __shared__ float smem[];
    float* sA   = smem;                // column-major: sA[j*D1 + i], 60000 floats
    float* sRed = smem + D1 * NN;      // 256-float reduction scratch

    const int tid  = threadIdx.x;
    const int lane = tid & 31;
    const int wave = tid >> 5;
    float* G = X1 + (size_t)blockIdx.x * D1 * NN;   // row-major [300][200]

    // Load + transpose to column-major (coalesced global reads)
    for (int idx = tid; idx < D1 * NN; idx += 256) {
        const int i = idx / NN, j = idx % NN;
        sA[j * D1 + i] = G[idx];
    }
    __syncthreads();

    for (int j = 0; j < NN; ++j) {
        float* qj = sA + j * D1;

        // ---- normalize column j (block-wide reduction) ----
        float p = 0.f;
        for (int i = tid; i < D1; i += 256) { const float v = qj[i]; p += v * v; }
        sRed[tid] = p;
        __syncthreads();
        for (int s = 128; s > 0; s >>= 1) {
            if (tid < s) sRed[tid] += sRed[tid + s];
            __syncthreads();
        }
        const float inv = rsqrtf(sRed[0] + 1e-30f);
        for (int i = tid; i < D1; i += 256) qj[i] *= inv;
        __syncthreads();

        // ---- orthogonalize columns j+1..N-1 against q_j (wave per column) ----
        for (int k = j + 1 + wave; k < NN; k += 8) {
            float* ak = sA + k * D1;
            float d = 0.f;
            for (int i = lane; i < D1; i += 32) d += qj[i] * ak[i];
            for (int off = 16; off > 0; off >>= 1) d += __shfl_down(d, off, 32);
            d = __shfl(d, 0, 32);
            for (int i = lane; i < D1; i += 32) ak[i] -= d * qj[i];
        }
        __syncthreads();
    }

    // Write Q back row-major, in place
    for (int idx = tid; idx < D1 * NN; idx += 256) {
        const int i = idx / NN, j = idx % NN;
        G[idx] = sA[j * D1 + i];
    }
}

// ---------------------------------------------------------------------------
// Kernel 3: logits = Q_flat @ fc_w + fc_b, fused log_softmax.
// One block per batch row; 10 register accumulators/thread; LDS tree reduce.
// Memory-bound: ~284 MB total traffic => ~12 us at 23.3 TB/s HBM.
// ---------------------------------------------------------------------------
__global__ __launch_bounds__(256)
void fc_logsoftmax(const float* __restrict__ Q, const float* __restrict__ W,
                   const float* __restrict__ bias, float* __restrict__ out)
{
    __shared__ float sAcc[NCLS * 256];
    __shared__ float sLog[NCLS];

    const int b   = blockIdx.x;
    const int tid = threadIdx.x;
    const float* q = Q + (size_t)b * FC_IN;

    float acc[NCLS];
#pragma unroll
    for (int c = 0; c < NCLS; ++c) acc[c] = 0.f;

    for (int i = tid; i < FC_IN; i += 256) {
        __builtin_prefetch(q + i + 4 * 256, 0, 3);
        const float v = q[i];
        const float* wr = W + (size_t)i * NCLS;
#pragma unroll
        for (int c = 0; c < NCLS; ++c) acc[c] = fmaf(v, wr[c], acc[c]);
    }
#pragma unroll
    for (int c = 0; c < NCLS; ++c) sAcc[c * 256 + tid] = acc[c];
    __syncthreads();
    for (int s = 128; s > 0; s >>= 1) {
        if (tid < s) {
#pragma unroll
            for (int c = 0; c < NCLS; ++c)
                sAcc[c * 256 + tid] += sAcc[c * 256 + tid + s];
        }
        __syncthreads();
    }
    if (tid < NCLS) sLog[tid] = sAcc[tid * 256] + bias[tid];
    __syncthreads();
    if (tid == 0) {
        float m = -INFINITY;
        for (int c = 0; c < NCLS; ++c) m = fmaxf(m, sLog[c]);
        float sum = 0.f;
        for (int c = 0; c < NCLS; ++c) sum += __expf(sLog[c] - m);
        const float lse = m + __logf(sum);
        for (int c = 0; c < NCLS; ++c) out[b * NCLS + c] = sLog[c] - lse;
    }
}

// ---------------------------------------------------------------------------
extern "C" void kernel_launch(void* const* d_in, const int* in_sizes, int n_in,
                              void* d_out, int out_size, void* d_ws, size_t ws_size,
                              hipStream_t stream)
{
    const float* x       = (const float*)d_in[0];  // [64,400,200]
    const float* filters = (const float*)d_in[1];  // [16,300,400]
    const float* fc_w    = (const float*)d_in[2];  // [960000,10]
    const float* fc_b    = (const float*)d_in[3];  // [1,10]
    float* X1 = (float*)d_ws;                      // [64,16,300,200] = 245.76 MB

    // 1) FRMap: 1024 GEMMs via fp32 WMMA
    dim3 g1(MT, BATCH * NF);
    frmap_wmma<<<g1, 128, 0, stream>>>(x, filters, X1);

    // 2) ReOrth: in-place MGS QR, whole matrix resident in LDS (240KB + scratch)
    const size_t ldsBytes = (size_t)(D1 * NN + 256) * sizeof(float);
    mgs_qr<<<BATCH * NF, 256, ldsBytes, stream>>>(X1);

    // 3) FC + log_softmax
    fc_logsoftmax<<<BATCH, 256, 0, stream>>>(X1, fc_w, fc_b, (float*)d_out);
}